// Self_Pressive_Network_82755429860026
// MI455X (gfx1250) — compile-verified
//
#include <hip/hip_runtime.h>
#include <hip/hip_bf16.h>
#include <math.h>

#define NN 4096
#define DD 128

typedef __attribute__((ext_vector_type(2))) float v2f;
typedef __attribute__((ext_vector_type(8))) float v8f;

__device__ __forceinline__ v8f wmma_f32(v2f a, v2f b, v8f c) {
  // V_WMMA_F32_16X16X4_F32 : D = A(16x4) * B(4x16) + C(16x16)
  return __builtin_amdgcn_wmma_f32_16x16x4_f32(false, a, false, b, (short)0, c,
                                               false, false);
}

// ---------------------------------------------------------------------------
// zero the 5 workspace accumulators (S is fully overwritten by tile_kernel,
// d_out[0] by final_kernel, so no large zero pass is needed)
// ---------------------------------------------------------------------------
__global__ __launch_bounds__(32) void zero_kernel(float* __restrict__ acc5) {
  if (threadIdx.x < 8) acc5[threadIdx.x] = 0.0f;
}

// ---------------------------------------------------------------------------
// express MLP: ez = tanh(relu(relu(z W0^T + b0) W1^T + b1) W2^T + b2)
// one block = 16 nodes of one modality, 8 waves (one 16-wide output tile each)
// ---------------------------------------------------------------------------
__global__ __launch_bounds__(256) void mlp_kernel(const float* __restrict__ zs,
                                                  const float* __restrict__ W,
                                                  const float* __restrict__ b,
                                                  float* __restrict__ ez) {
  const int m = blockIdx.x >> 8;  // 256 node-blocks per modality
  const int nb = blockIdx.x & 255;
  const int tid = threadIdx.x;
  const int wave = tid >> 5;
  const int L = tid & 31;
  const int c = L & 15;   // lane column
  const int h = L >> 4;   // lane half
  const int o0 = wave << 4;

  __shared__ float buf[2][16][132];  // padded: conflict-free b64 access

  // stage zs rows (16 x 128) into buf[0]
  {
    const float* src = zs + ((size_t)m * NN + (size_t)nb * 16) * DD;
    int row = tid >> 4;
    int col = (tid & 15) * 8;
#pragma unroll
    for (int q = 0; q < 8; ++q) buf[0][row][col + q] = src[row * DD + col + q];
  }
  __syncthreads();

  int cur = 0;
  for (int l = 0; l < 3; ++l) {
    const float* wrow = W + (((size_t)m * 3 + l) * DD + (o0 + c)) * DD;
    const float* bl = b + ((size_t)m * 3 + l) * DD;

    v8f acc = {0.f, 0.f, 0.f, 0.f, 0.f, 0.f, 0.f, 0.f};
#pragma unroll
    for (int ks = 0; ks < 32; ++ks) {
      const int k0 = 4 * ks + 2 * h;
      v2f a = *(const v2f*)&buf[cur][c][k0];   // A[r=c][k0..k0+1]
      v2f bb = *(const v2f*)(wrow + k0);       // B[k][o] = W[o][k]
      acc = wmma_f32(a, bb, acc);
    }

    const float bias = bl[o0 + c];
    float vals[8];
#pragma unroll
    for (int g = 0; g < 8; ++g) {
      float v = acc[g] + bias;
      vals[g] = (l < 2) ? fmaxf(v, 0.0f) : tanhf(v);
    }
#pragma unroll
    for (int g = 0; g < 8; ++g) buf[cur ^ 1][g + 8 * h][o0 + c] = vals[g];
    __syncthreads();
    cur ^= 1;
  }

  // write ez
  {
    float* dst = ez + ((size_t)m * NN + (size_t)nb * 16) * DD;
    int row = tid >> 4;
    int col = (tid & 15) * 8;
#pragma unroll
    for (int q = 0; q < 8; ++q) dst[row * DD + col + q] = buf[cur][row][col + q];
  }
}

// ---------------------------------------------------------------------------
// fully fused tile kernel, both modalities per workgroup:
//   gram(m=0,1) -> soft-threshold C tiles -> reg partials
//   -> S = b0*C0 + b1*C1 written with plain stores (no atomics, no pre-zero)
//   -> (S-K)^2, (S-P)^2 partials fused
//   -> Cz(m) accumulated via WMMA (C relayout D->A through per-wave LDS)
//   -> cross-wave Cz reduce in LDS -> rec_loss partial
// grid: 256 blocks (one per 16-row stripe), 128 threads = 4 waves,
// wave w handles j-tiles w, w+4, ...
// ---------------------------------------------------------------------------
__global__ __launch_bounds__(128) void tile_kernel(
    const float* __restrict__ zs, const float* __restrict__ ez,
    const float* __restrict__ thres_p, const float* __restrict__ beta_p,
    const float* __restrict__ Km, const float* __restrict__ Pm,
    float* __restrict__ S, float* __restrict__ acc5) {
  const int i0 = blockIdx.x * 16;
  const int tid = threadIdx.x;
  const int wave = tid >> 5;
  const int L = tid & 31;
  const int c = L & 15;
  const int h = L >> 4;

  const float thres = thres_p[0];
  const float b0 = beta_p[0];
  const float b1 = beta_p[1];
  const float shrink = 1.0f / 128.0f;

  __shared__ float cstage[4][2][16][20];  // per-wave D->A relay (padded)
  __shared__ float czbuf[2][16][130];     // cross-wave Cz accumulator (padded)
  __shared__ float red[5][4];

  // zero czbuf
  for (int t = tid; t < 2 * 16 * 130; t += 128) (&czbuf[0][0][0])[t] = 0.0f;
  __syncthreads();

  // preload A fragments for this stripe, both modalities (reused by all jb)
  v2f afr[2][32];
#pragma unroll
  for (int m = 0; m < 2; ++m) {
    const float* arow = ez + ((size_t)m * NN + i0 + c) * DD + 2 * h;
#pragma unroll
    for (int ks = 0; ks < 32; ++ks) afr[m][ks] = *(const v2f*)(arow + 4 * ks);
  }

  v8f cz[2][8];
#pragma unroll
  for (int m = 0; m < 2; ++m)
#pragma unroll
    for (int dt = 0; dt < 8; ++dt)
      cz[m][dt] = (v8f){0.f, 0.f, 0.f, 0.f, 0.f, 0.f, 0.f, 0.f};

  float reg_abs = 0.0f, reg_sq = 0.0f, l1 = 0.0f, l2 = 0.0f;

  for (int jb = wave; jb < 256; jb += 4) {
    const int j0 = jb * 16;

    // gram tiles for both modalities: 2 x 32 WMMA over K=128
    v8f g[2];
#pragma unroll
    for (int m = 0; m < 2; ++m) {
      g[m] = (v8f){0.f, 0.f, 0.f, 0.f, 0.f, 0.f, 0.f, 0.f};
      const float* brow = ez + ((size_t)m * NN + j0 + c) * DD + 2 * h;
#pragma unroll
      for (int ks = 0; ks < 32; ++ks) {
        v2f bfr = *(const v2f*)(brow + 4 * ks);
        g[m] = wmma_f32(afr[m][ks], bfr, g[m]);
      }
    }

    // soft-threshold, diag zero, reg partials, S store, fused (S-K)^2/(S-P)^2
#pragma unroll
    for (int gi = 0; gi < 8; ++gi) {
      const int row = i0 + gi + 8 * h;
      const int col = j0 + c;
      float cv[2];
#pragma unroll
      for (int m = 0; m < 2; ++m) {
        float gv = g[m][gi];
        float t = copysignf(fmaxf(fabsf(gv) - thres, 0.0f), gv) * shrink;
        if (row == col) t = 0.0f;
        reg_abs += fabsf(t);
        reg_sq += t * t;
        cstage[wave][m][gi + 8 * h][c] = t;
        cv[m] = t;
      }
      const float s = b0 * cv[0] + b1 * cv[1];
      const size_t off = (size_t)row * NN + col;
      S[off] = s;
      float d1 = s - Km[off];
      float d2 = s - Pm[off];
      l1 += d1 * d1;
      l2 += d2 * d2;
    }

    // Cz[m] += C_tile(16x16) @ zs[m, j0..j0+15, :]  (per-wave in-order LDS)
#pragma unroll
    for (int m = 0; m < 2; ++m) {
      const float* zmm = zs + (size_t)m * NN * DD;
#pragma unroll
      for (int kt = 0; kt < 4; ++kt) {
        v2f a = *(const v2f*)&cstage[wave][m][c][4 * kt + 2 * h];
        const float* zrow = zmm + (size_t)(j0 + 4 * kt + 2 * h) * DD + c;
#pragma unroll
        for (int dt = 0; dt < 8; ++dt) {
          v2f bfr;
          bfr.x = zrow[dt * 16];       // B[k  ][d]
          bfr.y = zrow[DD + dt * 16];  // B[k+1][d]
          cz[m][dt] = wmma_f32(a, bfr, cz[m][dt]);
        }
      }
    }
  }

  // cross-wave Cz reduction (LDS float atomics, conflict-free padding)
#pragma unroll
  for (int m = 0; m < 2; ++m)
#pragma unroll
    for (int dt = 0; dt < 8; ++dt)
#pragma unroll
      for (int gi = 0; gi < 8; ++gi)
        atomicAdd(&czbuf[m][gi + 8 * h][dt * 16 + c], cz[m][dt][gi]);
  __syncthreads();

  // rec_loss partial: sum (zs - Cz)^2 over this stripe, both modalities
  float rec_sq = 0.0f;
  for (int t = tid; t < 2 * 16 * 128; t += 128) {
    const int m = t >> 11;
    const int row = (t >> 7) & 15;
    const int col = t & 127;
    float zv = zs[((size_t)m * NN + i0 + row) * DD + col];
    float r = zv - czbuf[m][row][col];
    rec_sq += r * r;
  }

  // block reduction of the 5 scalar partials
  float v5[5] = {reg_abs, reg_sq, rec_sq, l1, l2};
#pragma unroll
  for (int q = 0; q < 5; ++q) {
#pragma unroll
    for (int off = 16; off >= 1; off >>= 1) v5[q] += __shfl_xor(v5[q], off, 32);
  }
  if (L == 0) {
#pragma unroll
    for (int q = 0; q < 5; ++q) red[q][wave] = v5[q];
  }
  __syncthreads();
  if (tid == 0) {
#pragma unroll
    for (int q = 0; q < 5; ++q) {
      float t = red[q][0] + red[q][1] + red[q][2] + red[q][3];
      atomicAdd(&acc5[q], t);
    }
  }
}

// ---------------------------------------------------------------------------
// final scalar loss
// ---------------------------------------------------------------------------
__global__ __launch_bounds__(32) void final_kernel(const float* __restrict__ acc5,
                                                   float* __restrict__ out) {
  if (threadIdx.x == 0) {
    const float LMBD = 0.9f, EXPRESS_L2 = 0.01f;
    float reg = LMBD * acc5[0] + 0.5f * (1.0f - LMBD) * acc5[1];
    float loss = (acc5[2] + reg * EXPRESS_L2 + acc5[3] + acc5[4]) / (float)NN;
    out[0] = loss;
  }
}

extern "C" void kernel_launch(void* const* d_in, const int* in_sizes, int n_in,
                              void* d_out, int out_size, void* d_ws,
                              size_t ws_size, hipStream_t stream) {
  const float* zs = (const float*)d_in[0];     // [2,4096,128]
  const float* W = (const float*)d_in[1];      // [2,3,128,128]
  const float* b = (const float*)d_in[2];      // [2,3,128]
  const float* thres = (const float*)d_in[3];  // [1]
  const float* beta = (const float*)d_in[4];   // [2]
  const float* Km = (const float*)d_in[5];     // [4096,4096]
  const float* Pm = (const float*)d_in[6];     // [4096,4096]

  float* out = (float*)d_out;  // [0]=loss, [1..]=S row-major
  float* Smat = out + 1;

  float* ez = (float*)d_ws;                // [2,4096,128] = 4 MB
  float* acc5 = ez + (size_t)2 * NN * DD;  // 5 accumulators

  zero_kernel<<<1, 32, 0, stream>>>(acc5);
  mlp_kernel<<<512, 256, 0, stream>>>(zs, W, b, ez);
  tile_kernel<<<256, 128, 0, stream>>>(zs, ez, thres, beta, Km, Pm, Smat, acc5);
  final_kernel<<<1, 32, 0, stream>>>(acc5, out);
}